// Decoder_7438883357110
// MI455X (gfx1250) — compile-verified
//
#include <hip/hip_runtime.h>
#include <math.h>

// Problem constants from the reference
#define BATCH 16
#define LAT   256
#define NPRIM 32
#define HH    256
#define WW    256
#define NOUT  (NPRIM * 5)   // 160 output cols of the decode GEMM

typedef __attribute__((ext_vector_type(2))) float v2f;
typedef __attribute__((ext_vector_type(8))) float v8f;

// ---------------------------------------------------------------------------
// Kernel 1: decode GEMM  raw[16,160] = inp[16,256] @ W[256,160] + b[160]
// One block of 10 wave32s; wave w owns output columns [16w, 16w+16).
// K=256 swept as 64 x V_WMMA_F32_16X16X4_F32 (full f32 precision, trivial cost).
//
// f32 16x4 A layout (ISA 7.12.2): lanes 0-15 -> K={0,1} in v0/v1,
// lanes 16-31 -> K={2,3}.  B (4x16) mirrors with N across lanes.
// C/D: vgpr r holds M=r (lanes 0-15) / M=r+8 (lanes 16-31), N=lane&15.
// ---------------------------------------------------------------------------
__global__ void __launch_bounds__(320)
decode_wmma(const float* __restrict__ inp,
            const float* __restrict__ Wm,
            const float* __restrict__ bias,
            float* __restrict__ raw) {
  const int lane  = threadIdx.x & 31;
  const int wave  = threadIdx.x >> 5;     // 0..9
  const int col0  = wave * 16;
  const int idx15 = lane & 15;            // row (A) / col (B,C)
  const int half  = lane >> 4;            // 0 or 1

#if defined(__gfx1250__) && __has_builtin(__builtin_amdgcn_wmma_f32_16x16x4_f32)
  v8f c = {0.f, 0.f, 0.f, 0.f, 0.f, 0.f, 0.f, 0.f};
  for (int k = 0; k < LAT; k += 4) {
    const int kk = k + 2 * half;
    v2f a, bf;
    a.x  = inp[idx15 * LAT + kk];
    a.y  = inp[idx15 * LAT + kk + 1];
    bf.x = Wm[(kk)     * NOUT + col0 + idx15];
    bf.y = Wm[(kk + 1) * NOUT + col0 + idx15];
    // 8 args: (neg_a, A, neg_b, B, c_mod, C, reuse_a, reuse_b)
    c = __builtin_amdgcn_wmma_f32_16x16x4_f32(false, a, false, bf,
                                              (short)0, c, false, false);
  }
#pragma unroll
  for (int r = 0; r < 8; ++r) {
    const int m = r + 8 * half;
    raw[m * NOUT + col0 + idx15] = c[r] + bias[col0 + idx15];
  }
#else
  // Scalar fallback (only taken if the f32x4 WMMA builtin is absent).
  for (int e = threadIdx.x; e < BATCH * NOUT; e += blockDim.x) {
    const int m = e / NOUT, n = e % NOUT;
    float acc = bias[n];
    for (int k = 0; k < LAT; ++k) acc += inp[m * LAT + k] * Wm[k * NOUT + n];
    raw[e] = acc;
  }
#endif
}

// ---------------------------------------------------------------------------
// Kernel 2: per-primitive parameter decode (512 threads total).
// Emits, per (b,n): {p1x, p1y, dx, dy, 1/len2, -log2(e)/sigma2}
// so the raster inner loop is pure fma + one v_exp_f32 (exp2 form).
// ---------------------------------------------------------------------------
__global__ void decode_params(const float* __restrict__ raw,
                              float* __restrict__ params) {
  const int t = blockIdx.x * blockDim.x + threadIdx.x;
  if (t >= BATCH * NPRIM) return;
  const int bb = t >> 5, nn = t & (NPRIM - 1);
  const float* r = raw + bb * NOUT + nn * 5;

  float p[4];
#pragma unroll
  for (int i = 0; i < 4; ++i) p[i] = 1.f / (1.f + __expf(-r[i]));

  const float x  = r[4];
  const float sp = (x > 20.f) ? x : log1pf(__expf(x));   // softplus
  const float s2 = sp * 1e-2f + 1e-4f;

  const float dx = p[2] - p[0], dy = p[3] - p[1];
  const float inv_len2 = 1.f / (dx * dx + dy * dy + 1e-12f);

  float* o = params + t * 6;
  o[0] = p[0];
  o[1] = p[1];
  o[2] = dx;
  o[3] = dy;
  o[4] = inv_len2;
  o[5] = -1.4426950408889634f / s2;   // exp(-d2/s2) == exp2(d2 * o[5])
}

// ---------------------------------------------------------------------------
// Kernel 3: raster. grid = (HW/256, B), one pixel per thread.
// 32-primitive params broadcast from LDS (conflict-free same-address reads).
// Compute-bound: ~14 VALU + 1 trans per (pixel, primitive).
// ---------------------------------------------------------------------------
__global__ void __launch_bounds__(256)
raster(const float* __restrict__ params,
       float* __restrict__ images,
       float* __restrict__ hard) {
  __shared__ float sp[NPRIM * 6];
  const int b = blockIdx.y;
  if (threadIdx.x < NPRIM * 6)
    sp[threadIdx.x] = params[b * NPRIM * 6 + threadIdx.x];
  __syncthreads();

  const int pix = blockIdx.x * blockDim.x + threadIdx.x;   // 0..65535
  const int py  = pix >> 8;
  const int px  = pix & (WW - 1);
  const float gx = px * (1.f / (WW - 1));
  const float gy = py * (1.f / (HH - 1));

  const float eps = 1.f / (HH - 1);
  const float nn2 = 2.f * (0.5f * eps) * (0.5f * eps);     // hard threshold

  float prod  = 1.f;    // running prod(1 - r)
  int   anyhit = 0;     // hard soft-OR == any hit

#pragma unroll 4
  for (int n = 0; n < NPRIM; ++n) {
    const float* q = sp + n * 6;
    const float rx = gx - q[0];
    const float ry = gy - q[1];
    const float dx = q[2], dy = q[3];
    float t = __builtin_fmaf(ry, dy, rx * dx) * q[4];
    t = fminf(fmaxf(t, 0.f), 1.f);                         // -> v_med3-style clamp
    const float ex = __builtin_fmaf(-t, dx, rx);
    const float ey = __builtin_fmaf(-t, dy, ry);
    const float d2 = __builtin_fmaf(ey, ey, ex * ex);
    const float e  = exp2f(d2 * q[5]);                     // exp(-d2/sigma2)
    prod = __builtin_fmaf(-e, prod, prod);                 // prod *= (1 - e)
    anyhit |= (d2 <= nn2) ? 1 : 0;
  }

  const int off = b * (HH * WW) + pix;
  images[off] = 1.f - prod;                                // CONTRAST == 1
  hard[off]   = anyhit ? 1.f : 0.f;
}

// ---------------------------------------------------------------------------
// Host-side launch
// ---------------------------------------------------------------------------
extern "C" void kernel_launch(void* const* d_in, const int* in_sizes, int n_in,
                              void* d_out, int out_size, void* d_ws, size_t ws_size,
                              hipStream_t stream) {
  (void)in_sizes; (void)n_in; (void)out_size; (void)ws_size;

  const float* inp  = (const float*)d_in[0];   // [16,256]
  const float* Wm   = (const float*)d_in[1];   // [256,160]
  const float* bias = (const float*)d_in[2];   // [160]

  float* raw    = (float*)d_ws;                // 16*160 f32
  float* params = raw + BATCH * NOUT;          // 16*32*6 f32

  float* images = (float*)d_out;               // [16,1,256,256]
  float* hard   = images + BATCH * HH * WW;    // [16,1,256,256]

  decode_wmma<<<1, 320, 0, stream>>>(inp, Wm, bias, raw);
  decode_params<<<1, BATCH * NPRIM, 0, stream>>>(raw, params);

  dim3 grid(HH * WW / 256, BATCH);
  raster<<<grid, 256, 0, stream>>>(params, images, hard);
}